// CPCircuitLayer_63350767616542
// MI455X (gfx1250) — compile-verified
//
#include <hip/hip_runtime.h>

typedef __attribute__((ext_vector_type(2))) float v2f;
typedef __attribute__((ext_vector_type(8))) float v8f;

#define H_DIM 512   // hidden size (= N = K of main GEMM)
#define R_DIM 64    // rank
#define M_ROWS 2048 // B*S rows

// ---------------------------------------------------------------------------
// Kernel 1: fuse the two small factors into M[hp][h] = sum_r W_seq[r,hp] *
//           hidden_embeddings[h,r] * cp_weight[r].   (512x512, 33 MFLOP)
// ---------------------------------------------------------------------------
__global__ void build_M(const float* __restrict__ W_seq,   // (R, H) row-major
                        const float* __restrict__ he,      // (H, R) row-major
                        const float* __restrict__ cp,      // (R)
                        float* __restrict__ Mmat) {        // (H, H): [hp][h]
    const int idx = blockIdx.x * blockDim.x + threadIdx.x; // 0 .. 512*512-1
    const int hp = idx >> 9;   // slow index: row of M  (k-dim of main GEMM)
    const int h  = idx & 511;  // fast index: col of M  (coalesced store)
    float acc = 0.0f;
#pragma unroll 8
    for (int r = 0; r < R_DIM; ++r) {
        // W_seq[r*H + hp] is uniform across the 512 consecutive threads of a
        // row-group (scalar-friendly); he[h*R + r] is small & L2-resident.
        acc = fmaf(W_seq[r * H_DIM + hp], he[h * R_DIM + r] * cp[r], acc);
    }
    Mmat[idx] = acc;
}

// ---------------------------------------------------------------------------
// Kernel 2: Out(2048x512) = X(2048x512) @ M(512x512), f32 WMMA.
// Each wave computes a 16x64 strip: 4 accumulators, A-tile reused 4x.
// K-loop steps 4 per v_wmma_f32_16x16x4_f32.
//
// VGPR layouts (ISA 7.12.2, wave32):
//   A 16x4  : lane l<16 -> M=l, a[0]=K0, a[1]=K1 ; lanes 16-31 -> K2,K3
//   B 4x16  : lane l<16 -> N=l, b[0]=row K0, b[1]=row K1 ; half2 -> K2,K3
//   C/D 16x16: vgpr p, lanes 0-15 -> M=p, N=l ; lanes 16-31 -> M=p+8
// ---------------------------------------------------------------------------
__global__ void __launch_bounds__(256)
gemm_out(const float* __restrict__ Xin,   // (2048, 512)
         const float* __restrict__ Mmat,  // (512, 512)
         float* __restrict__ Out) {       // (2048, 512)
    const int lane = threadIdx.x & 31;
    const int wave = threadIdx.x >> 5;
    const int l    = lane & 15;
    const int half = lane >> 4;

    // 1024 waves total: tile_m in [0,128), ngroup in [0,8)
    const int wid    = blockIdx.x * (blockDim.x >> 5) + wave;
    const int tile_m = wid >> 3;
    const int ngroup = wid & 7;
    const int m0 = tile_m * 16;
    const int n0 = ngroup * 64;

    v8f acc0 = {}, acc1 = {}, acc2 = {}, acc3 = {};

    const float* __restrict__ arow = Xin + (size_t)(m0 + l) * H_DIM;

#pragma unroll 4
    for (int k = 0; k < H_DIM; k += 4) {
        const int ka = k + 2 * half;           // this half-wave's K pair
        v2f a;
        a.x = arow[ka];                        // contiguous b64 per lane
        a.y = arow[ka + 1];

        const float* __restrict__ b0 = Mmat + (size_t)ka * H_DIM + n0 + l;
        const float* __restrict__ b1 = b0 + H_DIM;

        v2f b;
        b.x = b0[0];  b.y = b1[0];
        acc0 = __builtin_amdgcn_wmma_f32_16x16x4_f32(
            false, a, false, b, (short)0, acc0, false, false);
        b.x = b0[16]; b.y = b1[16];
        acc1 = __builtin_amdgcn_wmma_f32_16x16x4_f32(
            false, a, false, b, (short)0, acc1, false, false);
        b.x = b0[32]; b.y = b1[32];
        acc2 = __builtin_amdgcn_wmma_f32_16x16x4_f32(
            false, a, false, b, (short)0, acc2, false, false);
        b.x = b0[48]; b.y = b1[48];
        acc3 = __builtin_amdgcn_wmma_f32_16x16x4_f32(
            false, a, false, b, (short)0, acc3, false, false);
    }

    // Store 16x64 strip: coalesced 16-lane segments per accumulator row.
    float* __restrict__ orow = Out + n0 + l;
#pragma unroll
    for (int p = 0; p < 8; ++p) {
        const size_t m = (size_t)(m0 + p + half * 8);
        orow[m * H_DIM +  0] = acc0[p];
        orow[m * H_DIM + 16] = acc1[p];
        orow[m * H_DIM + 32] = acc2[p];
        orow[m * H_DIM + 48] = acc3[p];
    }
}

// ---------------------------------------------------------------------------
extern "C" void kernel_launch(void* const* d_in, const int* in_sizes, int n_in,
                              void* d_out, int out_size, void* d_ws, size_t ws_size,
                              hipStream_t stream) {
    const float* hidden = (const float*)d_in[0];  // (B,S,H) f32
    // d_in[1] = all_indices (int64): structurally n = s*H + h -> unused.
    const float* W_seq  = (const float*)d_in[2];  // (R,H) f32
    const float* he     = (const float*)d_in[3];  // (H,R) f32
    const float* cp     = (const float*)d_in[4];  // (1,R) f32
    float* out  = (float*)d_out;                  // (B,S,H) f32
    float* Mmat = (float*)d_ws;                   // 512*512*4 = 1 MB scratch

    (void)in_sizes; (void)n_in; (void)out_size; (void)ws_size;

    // Stage 1: M = W_seq^T @ (he * cp)^T   (512x512)
    build_M<<<(H_DIM * H_DIM) / 256, 256, 0, stream>>>(W_seq, he, cp, Mmat);

    // Stage 2: Out = X @ M via f32 WMMA. 1024 waves -> 128 blocks x 8 waves.
    gemm_out<<<128, 256, 0, stream>>>(hidden, Mmat, out);
}